// RegressionLoss_395136991460
// MI455X (gfx1250) — compile-verified
//
#include <hip/hip_runtime.h>
#include <hip/hip_bf16.h>

typedef unsigned int       u32;
typedef unsigned long long u64;
typedef u32 v4u __attribute__((ext_vector_type(4)));
typedef int v8i __attribute__((ext_vector_type(8)));
typedef int v4i __attribute__((ext_vector_type(4)));

#define TPB  256   // 8 wave32s per block
#define ILP  4     // anchors per thread
#define MAXM 64    // annotations per batch (reference: M = 64)

// ---------------------------------------------------------------------------
// Kernel 1: per-(batch, anchor-chunk) partial loss.
//   grid = (chunks, B), block = 256. Each thread owns ILP anchors.
//   Annotation tile (M x 3 f32) is DMA'd into LDS with the CDNA5 Tensor Data
//   Mover (tensor_load_to_lds + s_wait_tensorcnt), then preprocessed once.
// ---------------------------------------------------------------------------
__global__ __launch_bounds__(TPB) void loss_partial(
    const float* __restrict__ regressions,   // [B][A][2]
    const float* __restrict__ anchors,       // [A][2]
    const float* __restrict__ annotations,   // [B][M][3]
    float2*      __restrict__ partials,      // [B][chunks] (num, cnt)
    int A, int M, int chunks)
{
    __shared__ float  raw[MAXM * 3];   // raw annotations for this batch
    __shared__ float4 tab[MAXM];       // preprocessed {X1, X2, len, 0}
    __shared__ float  rednum[TPB / 32];
    __shared__ float  redcnt[TPB / 32];

    const int b   = blockIdx.y;
    const int tid = threadIdx.x;

    // ---- TDM DMA: annotations[b] (M*3 dwords) -> LDS, issued by wave 0 ----
    if ((tid >> 5) == 0) {
        const u64 gaddr = (u64)(const void*)(annotations + (u64)b * (M * 3));
        const u32 lds   = (u32)(u64)(void*)&raw[0];   // low 32 bits = LDS byte offset
        const u32 n     = (u32)(M * 3);               // dwords in dim0 (192)
        // D# group 0: count=1 (user), lds_addr, 57-bit global addr, type=2
        v4u g0 = { 1u,
                   lds,
                   (u32)(gaddr & 0xFFFFFFFFull),
                   (u32)((gaddr >> 32) & 0x1FFFFFFull) | 0x80000000u };
        // D# group 1: data_size=4B, tensor_dim0=n, tensor_dim1=1,
        //             tile_dim0=n, tile_dim1=1, tensor_dim0_stride=n
        v8i g1 = { (int)0x00020000u,     // [17:16] data_size = 2 (4 bytes)
                   (int)(n << 16),       // [63:48] tensor_dim0[15:0]
                   (int)(1u << 16),      // [95:80] tensor_dim1[15:0] = 1
                   (int)(n << 16),       // [127:112] tile_dim0
                   1,                    // [143:128] tile_dim1 = 1
                   (int)n,               // [191:160] tensor_dim0_stride[31:0]
                   0, 0 };
        v4i gz4 = { 0, 0, 0, 0 };        // groups 2/3 unused (2-D tile)
        v8i gz8 = { 0, 0, 0, 0, 0, 0, 0, 0 };
        __builtin_amdgcn_tensor_load_to_lds(g0, g1, gz4, gz4, gz8, 0);
        __builtin_amdgcn_s_wait_tensorcnt(0);
    }
    __syncthreads();

    // ---- preprocess annotations once: invalid -> IoU forced to 0 ----------
    if (tid < M) {
        const float x1 = raw[tid * 3 + 0];
        const float x2 = raw[tid * 3 + 1];
        const float lb = raw[tid * 3 + 2];
        const bool valid = (lb != -1.0f);
        tab[tid] = valid ? make_float4(x1, x2, x2 - x1, 0.0f)
                         : make_float4(3.0e38f, -3.0e38f, 1.0f, 0.0f);
    }
    __syncthreads();

    // ---- per-thread anchors -----------------------------------------------
    const int base = blockIdx.x * (TPB * ILP) + tid;

    float a0v[ILP], a1v[ILP], awv[ILP], biv[ILP], buv[ILP];
    int   bjv[ILP];
    bool  actv[ILP];
#pragma unroll
    for (int k = 0; k < ILP; ++k) {
        const int idx = base + k * TPB;
        actv[k] = (idx < A);
        const float2 ap = actv[k] ? ((const float2*)anchors)[idx]
                                  : make_float2(0.0f, 0.0f);
        a0v[k] = ap.x;  a1v[k] = ap.y;  awv[k] = ap.y - ap.x;
        biv[k] = -1.0f; buv[k] = 1.0f;  bjv[k] = 0;      // iou = -1 seed
    }
    // gfx1250 stream prefetch of this block's regression rows
    if (base < A)
        __builtin_prefetch(&regressions[((u64)b * A + base) * 2], 0, 0);

    // ---- IoU max/argmax, division-free (track best as (iw, ua) pair) ------
    for (int j = 0; j < M; ++j) {
        const float4 an = tab[j];                 // one ds_load_b128, broadcast
#pragma unroll
        for (int k = 0; k < ILP; ++k) {
            const float tmin = fminf(a1v[k], an.y);
            const float tmax = fmaxf(a0v[k], an.x);
            const float iw   = fmaxf(tmin - tmax, 0.0f);
            const float ua   = fmaxf(awv[k] + an.z - iw, 1e-8f);
            // iw/ua > bi/bu  <=>  iw*bu > bi*ua   (ua, bu > 0)
            const bool upd = (iw * buv[k]) > (biv[k] * ua);
            biv[k] = upd ? iw : biv[k];
            buv[k] = upd ? ua : buv[k];
            bjv[k] = upd ? j  : bjv[k];
        }
    }

    // ---- smooth-L1 for positive anchors -----------------------------------
    float num = 0.0f, cnt = 0.0f;
#pragma unroll
    for (int k = 0; k < ILP; ++k) {
        const int idx = base + k * TPB;
        const bool pos = actv[k] && (biv[k] >= 0.5f * buv[k]);   // iou_max >= 0.5
        if (pos) {
            const float gx1 = raw[bjv[k] * 3 + 0];
            const float gx2 = raw[bjv[k] * 3 + 1];
            float gw  = gx2 - gx1;
            const float gcx = gx1 + 0.5f * gw;
            gw = fmaxf(gw, 1.0f);
            const float aw  = awv[k];
            const float acx = a0v[k] + 0.5f * aw;
            const float tdx = ((gcx - acx) / aw) * 10.0f;        // / 0.1
            const float tdw = logf(gw / aw) * 5.0f;              // / 0.2
            const float2 rg = ((const float2*)regressions)[(u64)b * A + idx];
            const float d0 = fabsf(tdx - rg.x);
            const float d1 = fabsf(tdw - rg.y);
            const float th = 1.0f / 9.0f;
            const float s0 = (d0 <= th) ? 4.5f * d0 * d0 : d0 - (0.5f / 9.0f);
            const float s1 = (d1 <= th) ? 4.5f * d1 * d1 : d1 - (0.5f / 9.0f);
            num += s0 + s1;
            cnt += 1.0f;
        }
    }

    // ---- block reduction (wave32 shuffles, then cross-wave via LDS) -------
    for (int off = 16; off > 0; off >>= 1) {
        num += __shfl_down(num, off, 32);
        cnt += __shfl_down(cnt, off, 32);
    }
    const int lane = tid & 31, wid = tid >> 5;
    if (lane == 0) { rednum[wid] = num; redcnt[wid] = cnt; }
    __syncthreads();
    if (wid == 0) {
        num = (lane < TPB / 32) ? rednum[lane] : 0.0f;
        cnt = (lane < TPB / 32) ? redcnt[lane] : 0.0f;
        for (int off = 4; off > 0; off >>= 1) {
            num += __shfl_down(num, off, 32);
            cnt += __shfl_down(cnt, off, 32);
        }
        if (lane == 0)
            partials[(u64)b * chunks + blockIdx.x] = make_float2(num, cnt);
    }
}

// ---------------------------------------------------------------------------
// Kernel 2: deterministic final reduction. One block; wave w owns batch w.
// ---------------------------------------------------------------------------
__global__ __launch_bounds__(256) void loss_final(
    const float2* __restrict__ partials, float* __restrict__ out,
    int B, int chunks)
{
    __shared__ float pb[32];
    const int tid = threadIdx.x, lane = tid & 31, w = tid >> 5;
    const int nw = blockDim.x >> 5;
    for (int b = w; b < B; b += nw) {
        float num = 0.0f, cnt = 0.0f;
        for (int j = lane; j < chunks; j += 32) {
            const float2 p = partials[(u64)b * chunks + j];
            num += p.x; cnt += p.y;
        }
        for (int off = 16; off > 0; off >>= 1) {
            num += __shfl_down(num, off, 32);
            cnt += __shfl_down(cnt, off, 32);
        }
        if (lane == 0) {
            const float c2 = 2.0f * cnt;
            pb[b] = (c2 > 0.0f) ? num / fmaxf(c2, 1.0f) : 0.0f;
        }
    }
    __syncthreads();
    if (tid == 0) {
        float s = 0.0f;
        for (int b = 0; b < B; ++b) s += pb[b];
        out[0] = s / (float)B;
    }
}

// ---------------------------------------------------------------------------
extern "C" void kernel_launch(void* const* d_in, const int* in_sizes, int n_in,
                              void* d_out, int out_size, void* d_ws, size_t ws_size,
                              hipStream_t stream) {
    const float* regressions = (const float*)d_in[0];   // [B][A][2] f32
    const float* anchors     = (const float*)d_in[1];   // [1][A][2] f32
    const float* annotations = (const float*)d_in[2];   // [B][M][3] f32
    float*       out         = (float*)d_out;

    const int A = in_sizes[1] / 2;                      // 100000
    const int B = in_sizes[0] / (2 * A);                // 8
    const int M = in_sizes[2] / (3 * B);                // 64

    const int anchorsPerBlock = TPB * ILP;              // 1024
    const int chunks = (A + anchorsPerBlock - 1) / anchorsPerBlock;  // 98
    float2* partials = (float2*)d_ws;                   // B*chunks*8 B ≈ 6.3 KB

    dim3 grid(chunks, B);
    loss_partial<<<grid, TPB, 0, stream>>>(regressions, anchors, annotations,
                                           partials, A, M, chunks);
    loss_final<<<1, 256, 0, stream>>>(partials, out, B, chunks);
}